// RerankLoss_63350767616715
// MI455X (gfx1250) — compile-verified
//
#include <hip/hip_runtime.h>

typedef __attribute__((ext_vector_type(2))) float v2f;
typedef __attribute__((ext_vector_type(8))) float v8f;

#define MARGIN   1.0f
#define L        200
#define LP       208            // pad to multiple of 16
#define NTHREADS 128            // 4 waves (wave32)
#define NWAVES   (NTHREADS / 32)

// One block per sample. Computes per-sample mean hinge over pos/neg pairs.
//
// Mask folding: masked hinge = max(0, ci + oj) with
//   ci = (lab_i==1) ? MARGIN - o_i : -inf     (positives / padding killed)
//   oj = (lab_j==0) ? o_j          : -inf     (negatives / padding killed)
// -inf + finite = -inf, -inf + -inf = -inf, relu(-inf) = +0.0 exactly.
__global__ __launch_bounds__(NTHREADS)
void rerank_rows(const float* __restrict__ outp,
                 const int*   __restrict__ lab,
                 float*       __restrict__ per_sample) {
    __shared__ alignas(16) float sh_oj[LP];   // masked negative scores
    __shared__ alignas(16) float sh_ci[LP];   // masked (MARGIN - o_i)
    __shared__ int   sh_np;
    __shared__ float sh_red[NTHREADS];

    const int b   = blockIdx.x;
    const int tid = threadIdx.x;
    const float NEG_INF = -__builtin_inff();

    if (tid == 0) sh_np = 0;
    __syncthreads();

    // Stage row into LDS with -inf masking (also kills padding).
    int np_local = 0;
    for (int idx = tid; idx < LP; idx += NTHREADS) {
        float oj = NEG_INF;
        float ci = NEG_INF;
        if (idx < L) {
            const float o = outp[b * L + idx];
            const int   l = lab[b * L + idx];
            np_local += l;
            if (l == 0) oj = o;               // valid negative
            else        ci = MARGIN - o;      // valid positive
        }
        sh_oj[idx] = oj;
        sh_ci[idx] = ci;
    }
    if (np_local) atomicAdd(&sh_np, np_local);  // integer add: order-independent
    __syncthreads();

    const int lane = tid & 31;
    const int wave = tid >> 5;
    const int m    = lane & 15;          // A-matrix row M
    const int koff = (lane >> 4) * 2;    // A-matrix K: lanes 0-15 -> K=0,1; 16-31 -> K=2,3

    v8f d = {0.f, 0.f, 0.f, 0.f, 0.f, 0.f, 0.f, 0.f};
    v2f bones;
    bones.x = 1.0f;
    bones.y = 1.0f;

    // i-tiles distributed over waves (wave-uniform -> EXEC all-1s for WMMA).
    for (int it = wave; it < LP / 16; it += NWAVES) {
        const float ci = sh_ci[it * 16 + m];

        for (int jb = 0; jb < LP; jb += 4) {
            // 8B broadcast load: {oj, oj+1}, j = jb + koff (same addr per half-wave)
            const v2f q = *(const v2f*)(&sh_oj[jb + koff]);
            v2f a;
            a.x = fmaxf(ci + q.x, 0.0f);     // hinge at (i, j)   -> K = koff
            a.y = fmaxf(ci + q.y, 0.0f);     // hinge at (i, j+1) -> K = koff+1
            // D[m][n] += sum_k A[m][k] * 1 : matrix-pipe pair-sum reduction
            d = __builtin_amdgcn_wmma_f32_16x16x4_f32(
                    /*neg_a=*/false, a, /*neg_b=*/false, bones,
                    /*c_mod=*/(short)0, d, /*reuse_a=*/false, /*reuse_b=*/false);
        }
    }

    // D columns replicate row-sums: grand total = (sum of all D entries)/16.
    float partial = d[0] + d[1] + d[2] + d[3] + d[4] + d[5] + d[6] + d[7];
    sh_red[tid] = partial;
    __syncthreads();
    for (int s = NTHREADS / 2; s > 0; s >>= 1) {
        if (tid < s) sh_red[tid] += sh_red[tid + s];
        __syncthreads();
    }

    if (tid == 0) {
        const float total  = sh_red[0] * (1.0f / 16.0f);
        const int   np     = sh_np;
        const float npairs = (float)np * (float)(L - np);
        per_sample[b] = (npairs > 0.0f) ? (total / npairs) : 0.0f;
    }
}

// Deterministic single-block reduction of per-sample losses -> scalar mean.
__global__ __launch_bounds__(256)
void reduce_final(const float* __restrict__ ps, float* __restrict__ outv, int B) {
    __shared__ float sh[256];
    float s = 0.0f;
    for (int i = threadIdx.x; i < B; i += 256) s += ps[i];  // fixed order per thread
    sh[threadIdx.x] = s;
    __syncthreads();
    for (int st = 128; st > 0; st >>= 1) {
        if ((int)threadIdx.x < st) sh[threadIdx.x] += sh[threadIdx.x + st];
        __syncthreads();
    }
    if (threadIdx.x == 0) outv[0] = sh[0] / (float)B;
}

extern "C" void kernel_launch(void* const* d_in, const int* in_sizes, int n_in,
                              void* d_out, int out_size, void* d_ws, size_t ws_size,
                              hipStream_t stream) {
    const float* output = (const float*)d_in[0];
    const int*   labels = (const int*)d_in[1];
    const int    B      = in_sizes[0] / L;

    float* per_sample = (float*)d_ws;  // B floats of scratch

    rerank_rows<<<B, NTHREADS, 0, stream>>>(output, labels, per_sample);
    reduce_final<<<1, 256, 0, stream>>>(per_sample, (float*)d_out, B);
}